// GCNLayer_12584254177714
// MI455X (gfx1250) — compile-verified
//
#include <hip/hip_runtime.h>

typedef float v2f __attribute__((ext_vector_type(2)));
typedef float v8f __attribute__((ext_vector_type(8)));

#define DFEAT 128   // D_IN == D_OUT == 128
#define STRIPES 8   // row-stripes (16 rows each) processed per block

// ---------------------------------------------------------------------------
// Phase 1: h = x @ W using V_WMMA_F32_16X16X4_F32.
// Block = 256 threads = 8 waves; wave w owns output columns [16w, 16w+16).
// Each block processes 8 row-stripes (128 rows) while keeping its B tile
// (128x16 fp32 = 64 VGPRs/lane) resident in registers, so W is read from L2
// exactly once per block. Inner loop: 32 fully-unrolled WMMAs (K=4 each).
// ---------------------------------------------------------------------------
__global__ void __launch_bounds__(256)
gcn_gemm_wmma_f32(const float* __restrict__ x, const float* __restrict__ W,
                  float* __restrict__ h, int N) {
  const int wave = threadIdx.x >> 5;    // 0..7 -> column tile
  const int lane = threadIdx.x & 31;
  const int half = lane >> 4;           // 0: lanes 0-15, 1: lanes 16-31
  const int m    = lane & 15;
  const int col  = wave * 16 + m;

  // Preload B tile: B 4x16 fp32 layout -> lane(n + 16*half), VGPR j holds
  // B[4s + 2*half + j][n]. Two strided b32 loads per k-step.
  v2f breg[32];
#pragma unroll
  for (int s = 0; s < 32; ++s) {
    const int k = 4 * s + 2 * half;
    breg[s].x = W[(size_t)k * DFEAT + col];
    breg[s].y = W[(size_t)(k + 1) * DFEAT + col];
  }

  const int blockRow0 = blockIdx.x * (16 * STRIPES);
#pragma unroll 1
  for (int t = 0; t < STRIPES; ++t) {
    const int rowBase = blockRow0 + 16 * t;
    if (rowBase >= N) break;

    // A 16x4 fp32 layout: lane(m + 16*half), VGPR j holds A[m][2*half + j]
    // -> contiguous float2 per k-step (single global_load_b64).
    const int arow = (rowBase + m < N) ? (rowBase + m) : (N - 1);
    const float* __restrict__ Arow = x + (size_t)arow * DFEAT;

    v8f c = {};
#pragma unroll
    for (int s = 0; s < 32; ++s) {
      const int k = 4 * s + 2 * half;
      v2f a;
      a.x = Arow[k];
      a.y = Arow[k + 1];
      c = __builtin_amdgcn_wmma_f32_16x16x4_f32(
          /*neg_a=*/false, a, /*neg_b=*/false, breg[s],
          /*c_mod=*/(short)0, c, /*reuse_a=*/false, /*reuse_b=*/false);
    }

    // C/D 16x16 fp32 layout: VGPR r -> row (8*half + r), col = m.
    float* __restrict__ Hp = h + (size_t)(rowBase + half * 8) * DFEAT + col;
    if (rowBase + 16 <= N) {
      // Fast path: full stripe, unconditional stores (no exec-mask churn).
#pragma unroll
      for (int r = 0; r < 8; ++r) Hp[(size_t)r * DFEAT] = c[r];
    } else {
      // Ragged final stripe only.
#pragma unroll
      for (int r = 0; r < 8; ++r)
        if (rowBase + half * 8 + r < N) Hp[(size_t)r * DFEAT] = c[r];
    }
  }
}

// ---------------------------------------------------------------------------
// Phase 2a: out[n, d] = bias[d]   (d_out is poisoned by the harness)
// ---------------------------------------------------------------------------
__global__ void __launch_bounds__(256)
gcn_init_out(float* __restrict__ out, const float* __restrict__ bias,
             long long total) {
  const long long i = (long long)blockIdx.x * blockDim.x + threadIdx.x;
  if (i < total) out[i] = bias[i & (DFEAT - 1)];
}

// ---------------------------------------------------------------------------
// Phase 2b: scatter-add  out[dst] += h[src] * w  over 600K edges.
// 32 threads (one wave) per edge -> index/weight loads are wave-uniform and
// coalesce to a single request. Each thread moves one float4 (16B) and issues
// four hardware global_atomic_add_f32 (unsafeAtomicAdd). h and out (51 MB
// each) are resident in the 192 MB L2, so the random gather + atomic RMW
// traffic stays on-chip; HBM floor for this phase is ~620 MB ≈ 27 us.
// ---------------------------------------------------------------------------
__global__ void __launch_bounds__(256)
gcn_scatter(const float* __restrict__ h, const long long* __restrict__ src_idx,
            const long long* __restrict__ dst_idx, const float* __restrict__ ea,
            float* __restrict__ out, long long E) {
  const long long t = (long long)blockIdx.x * blockDim.x + threadIdx.x;
  const long long e = t >> 5;
  if (e >= E) return;
  const int chunk = (int)(t & 31) * 4;  // this thread's 4 columns

  const long long s = src_idx[e];
  const long long d = dst_idx[e];
  const float w = ea[e];

  const float4 v = *reinterpret_cast<const float4*>(h + s * DFEAT + chunk);
  float* o = out + d * DFEAT + chunk;
  unsafeAtomicAdd(o + 0, v.x * w);
  unsafeAtomicAdd(o + 1, v.y * w);
  unsafeAtomicAdd(o + 2, v.z * w);
  unsafeAtomicAdd(o + 3, v.w * w);
}

// ---------------------------------------------------------------------------
// Host-side launch. Inputs (setup_inputs order):
//   0: x          [N, 128] f32
//   1: edge_index [2, E]   i64
//   2: edge_attr  [E]      f32
//   3: kernel     [128,128] f32
//   4: bias       [128]    f32
// Output: [N, 128] f32. Workspace holds h ([N,128] f32, 51.2 MB).
// ---------------------------------------------------------------------------
extern "C" void kernel_launch(void* const* d_in, const int* in_sizes, int n_in,
                              void* d_out, int out_size, void* d_ws, size_t ws_size,
                              hipStream_t stream) {
  const float*     x    = (const float*)d_in[0];
  const long long* ei   = (const long long*)d_in[1];
  const float*     ea   = (const float*)d_in[2];
  const float*     W    = (const float*)d_in[3];
  const float*     bias = (const float*)d_in[4];

  const int       N = in_sizes[0] / DFEAT;
  const long long E = (long long)in_sizes[2];

  float* h   = (float*)d_ws;
  float* out = (float*)d_out;

  // 1) Dense projection via WMMA (B tiles register-resident per block).
  const int rowsPerBlock = 16 * STRIPES;
  const int nBlocks = (N + rowsPerBlock - 1) / rowsPerBlock;
  gcn_gemm_wmma_f32<<<nBlocks, 256, 0, stream>>>(x, W, h, N);

  // 2) Initialize output with bias.
  const long long total = (long long)N * DFEAT;
  const int initBlocks = (int)((total + 255) / 256);
  gcn_init_out<<<initBlocks, 256, 0, stream>>>(out, bias, total);

  // 3) Edge scatter with hardware fp32 atomics.
  const long long sthreads = E * 32;
  const int sblocks = (int)((sthreads + 255) / 256);
  gcn_scatter<<<sblocks, 256, 0, stream>>>(h, ei, ei + E, ea, out, E);
}